// RGCNLayer_80942953660966
// MI455X (gfx1250) — compile-verified
//
#include <hip/hip_runtime.h>

#define N_NODES  50000
#define IN_FEAT  128
#define OUT_FEAT 128
#define NUM_RELS 8
#define LDS_STRIDE 130   // even padding: 8B-aligned float2 LDS reads, no bank-aliased staging

typedef __attribute__((ext_vector_type(2))) float v2f;
typedef __attribute__((ext_vector_type(8))) float v8f;

// ---------------------------------------------------------------------------
// Kernel 0: zero the output accumulator (d_out is poisoned by the harness)
// ---------------------------------------------------------------------------
__global__ __launch_bounds__(256) void rgcn_zero_kernel(float4* __restrict__ out, int n4) {
    int i = blockIdx.x * 256 + threadIdx.x;
    if (i < n4) out[i] = make_float4(0.f, 0.f, 0.f, 0.f);
}

// ---------------------------------------------------------------------------
// Kernel 1: hw[r][n][o] = sum_i x[n][i] * W[r][i][o]   via V_WMMA_F32_16X16X4_F32
// grid = (ceil(3125/8), NUM_RELS), block = 256 (8 waves)
// Each block: stage W[r] transposed (Wt[o][i]) into LDS, then each wave computes
// a 16(M) x 128(O) strip: 8 accumulator tiles, K-loop of 32 steps of 4.
// ---------------------------------------------------------------------------
__global__ __launch_bounds__(256) void rgcn_gemm_kernel(
    const float* __restrict__ x, const float* __restrict__ w,
    float* __restrict__ hw)
{
    __shared__ float wt[IN_FEAT * LDS_STRIDE];   // 128*130*4 = 66,560 B (WGP has 320KB)

    const int r = blockIdx.y;
    const float* wr = w + (size_t)r * IN_FEAT * OUT_FEAT;

    // Stage W[r] transposed: wt[o][i] = W[r][i][o]. Coalesced global reads over o;
    // LDS write addr = o*130 + i -> consecutive o advances bank by 2 (conflict-light).
    for (int idx = threadIdx.x; idx < IN_FEAT * OUT_FEAT; idx += 256) {
        int k = idx >> 7;      // i (input feature)
        int n = idx & 127;     // o (output feature)
        wt[n * LDS_STRIDE + k] = wr[idx];
    }
    __syncthreads();

    const int wave  = threadIdx.x >> 5;
    const int lane  = threadIdx.x & 31;
    const int mtile = blockIdx.x * 8 + wave;
    if (mtile * 16 >= N_NODES) return;           // 50000/16 = 3125 exact tiles

    const int m0    = mtile * 16;
    const int lmod  = lane & 15;
    const int khalf = (lane >> 4) * 2;           // lanes 0-15: K={0,1}; 16-31: K={2,3}

    v8f acc[8] = {};

    // A fragment source: row m0+lmod, K pair starting at khalf (8B aligned)
    const float* arow = x + (size_t)(m0 + lmod) * IN_FEAT + khalf;

    for (int k0 = 0; k0 < IN_FEAT; k0 += 4) {
        v2f a = *(const v2f*)(arow + k0);
#pragma unroll
        for (int t = 0; t < 8; ++t) {
            // B fragment: B[k][n] = Wt[n][k]; lane needs K pair at fixed n = t*16+lmod
            v2f b = *(const v2f*)(&wt[(t * 16 + lmod) * LDS_STRIDE + k0 + khalf]);
            acc[t] = __builtin_amdgcn_wmma_f32_16x16x4_f32(
                false, a, false, b, (short)0, acc[t], false, false);
        }
    }

    // C/D layout: VGPR j -> M = j (lanes 0-15) / M = 8+j (lanes 16-31), N = lane%16
    float* orow = hw + (size_t)r * N_NODES * OUT_FEAT;
    const int rbase = m0 + (lane >> 4) * 8;
#pragma unroll
    for (int t = 0; t < 8; ++t) {
#pragma unroll
        for (int j = 0; j < 8; ++j) {
            orow[(size_t)(rbase + j) * OUT_FEAT + t * 16 + lmod] = acc[t][j];
        }
    }
}

// ---------------------------------------------------------------------------
// Kernel 2: per-edge gather + scatter-add.  One wave per edge.
// ---------------------------------------------------------------------------
__device__ __forceinline__ void gatomic_fadd(float* p, float v) {
    // Native fp32 global atomic add (no return -> STOREcnt), avoids CAS loop.
    asm volatile("global_atomic_add_f32 %0, %1, off" :: "v"(p), "v"(v) : "memory");
}

__global__ __launch_bounds__(256) void rgcn_scatter_kernel(
    const float* __restrict__ hw, const float* __restrict__ norm,
    const int* __restrict__ src, const int* __restrict__ dst,
    const int* __restrict__ rel, float* __restrict__ out, int n_edges)
{
    const int wid  = (blockIdx.x * 256 + threadIdx.x) >> 5;   // edge id
    if (wid >= n_edges) return;
    const int lane = threadIdx.x & 31;

    const int   s   = src[wid];
    const int   d   = dst[wid];
    const int   r   = rel[wid];
    const float nrm = norm[wid];

    // 512B coalesced gather per wave: lane covers outputs [lane*4, lane*4+3]
    const float4 v = *(const float4*)(hw + ((size_t)r * N_NODES + (size_t)s) * OUT_FEAT
                                      + (size_t)lane * 4);
    float* op = out + (size_t)d * OUT_FEAT + lane * 4;
    gatomic_fadd(op + 0, v.x * nrm);
    gatomic_fadd(op + 1, v.y * nrm);
    gatomic_fadd(op + 2, v.z * nrm);
    gatomic_fadd(op + 3, v.w * nrm);
}

// ---------------------------------------------------------------------------
// Kernel 3: h = relu(h + bias), in place
// ---------------------------------------------------------------------------
__global__ __launch_bounds__(256) void rgcn_bias_relu_kernel(
    float* __restrict__ out, const float* __restrict__ bias, int n)
{
    int i = blockIdx.x * 256 + threadIdx.x;
    if (i < n) {
        float v = out[i] + bias[i & (OUT_FEAT - 1)];
        out[i] = v > 0.f ? v : 0.f;
    }
}

// ---------------------------------------------------------------------------
extern "C" void kernel_launch(void* const* d_in, const int* in_sizes, int n_in,
                              void* d_out, int out_size, void* d_ws, size_t ws_size,
                              hipStream_t stream) {
    const float* x      = (const float*)d_in[0];   // [N, 128]
    const float* norm   = (const float*)d_in[1];   // [E, 1]
    const float* weight = (const float*)d_in[2];   // [R, 128, 128]
    const float* bias   = (const float*)d_in[3];   // [128]
    const int*   src    = (const int*)d_in[4];     // [E]
    const int*   dst    = (const int*)d_in[5];     // [E]
    const int*   rel    = (const int*)d_in[6];     // [E]
    float*       out    = (float*)d_out;           // [N, 128]
    float*       hw     = (float*)d_ws;            // [R, N, 128] = 204.8 MB scratch

    const int n_edges = in_sizes[4];
    const int n_out   = N_NODES * OUT_FEAT;        // 6.4M

    // 1) zero output accumulator
    rgcn_zero_kernel<<<(n_out / 4 + 255) / 256, 256, 0, stream>>>((float4*)out, n_out / 4);

    // 2) 8 GEMMs via fp32 WMMA: hw[r] = x @ W[r]
    dim3 ggrid((N_NODES / 16 + 7) / 8, NUM_RELS);  // (391, 8)
    rgcn_gemm_kernel<<<ggrid, 256, 0, stream>>>(x, weight, hw);

    // 3) edge gather * norm, scatter-add into out (one wave per edge)
    int sblocks = (n_edges * 32 + 255) / 256;
    rgcn_scatter_kernel<<<sblocks, 256, 0, stream>>>(hw, norm, src, dst, rel, out, n_edges);

    // 4) relu(out + bias)
    rgcn_bias_relu_kernel<<<(n_out + 255) / 256, 256, 0, stream>>>(out, bias, n_out);
}